// XSimGCL_Encoder_21371757264955
// MI455X (gfx1250) — compile-verified
//
#include <hip/hip_runtime.h>
#include <stdint.h>

// ---------------- problem constants (from reference) ----------------
#define U_CNT   100000
#define I_CNT   50000
#define D_DIM   64
#define N_CNT   (U_CNT + I_CNT)           // 150000
#define NNZ_CNT 4000000
#define EPS_F   0.2f
#define ND      ((long long)N_CNT * D_DIM) // 9,600,000 floats

#define EDGE_TILE 2048                     // edges per LDS tile
#define TILES_PER_BLOCK 2                  // double-buffered TDM

typedef unsigned int v4u __attribute__((ext_vector_type(4)));
typedef int          v4i __attribute__((ext_vector_type(4)));
typedef int          v8i __attribute__((ext_vector_type(8)));

#if __has_builtin(__builtin_amdgcn_tensor_load_to_lds)
#define HAVE_TDM 1
#else
#define HAVE_TDM 0
#endif

#if HAVE_TDM
// Issue one TDM 1-D copy: count 4-byte elements from global -> LDS.
// D# per CDNA5 ISA ch.8: group0 = {count=1|type=2|lds_addr|global_addr},
// group1 = {data_size=4B, tensor_dim0=count, tensor_dim1=1, tile_dim0=count,
//           tile_dim1=1, stride0=count}. Groups 2/3 unused (<=2D tensor).
__device__ __forceinline__ void tdm_load_1d(unsigned lds_byte_off,
                                            const void* gptr, unsigned count) {
  unsigned long long ga = (unsigned long long)gptr;
  v4u g0;
  g0.x = 1u;                                                  // valid-descriptor count = 1
  g0.y = lds_byte_off;                                        // lds_addr (bytes)
  g0.z = (unsigned)(ga & 0xFFFFFFFFull);                      // global_addr[31:0]
  g0.w = (unsigned)((ga >> 32) & 0x01FFFFFFull) | (2u << 30); // global_addr[56:32] | type=2
  v8i g1;
  g1[0] = (int)(2u << 16);                                    // data_size = 2 (4 bytes)
  g1[1] = (int)((count & 0xFFFFu) << 16);                     // tensor_dim0[15:0]
  g1[2] = (int)((count >> 16) | (1u << 16));                  // tensor_dim0[31:16], tensor_dim1=1
  g1[3] = (int)((count & 0xFFFFu) << 16);                     // tile_dim0 = count (<= 2048)
  g1[4] = 1;                                                  // tile_dim1 = 1, tile_dim2 = 0
  g1[5] = (int)count;                                         // tensor_dim0_stride[31:0]
  g1[6] = 0;
  g1[7] = 0;
  v4i gz = {0, 0, 0, 0};
#if defined(__clang_major__) && (__clang_major__ >= 23)
  v8i gz8 = {0, 0, 0, 0, 0, 0, 0, 0};
  __builtin_amdgcn_tensor_load_to_lds(g0, g1, gz, gz, gz8, 0);
#else
  __builtin_amdgcn_tensor_load_to_lds(g0, g1, gz, gz, 0);
#endif
}
#endif // HAVE_TDM

__device__ __forceinline__ unsigned lds_off(const void* p) {
  return (unsigned)(unsigned long long)p;   // flat shared addr: low 32 bits = LDS offset
}

// ---------------- kernel 1: build ego0, zero first accumulator ----------------
__global__ void __launch_bounds__(256)
init_ego(const float* __restrict__ ue, const float* __restrict__ ie,
         float* __restrict__ A, float* __restrict__ B) {
  const long long i = (long long)blockIdx.x * 256 + threadIdx.x;   // float4 index
  const long long tot4 = ND / 4;
  if (i >= tot4) return;
  const long long ud4 = (long long)U_CNT * D_DIM / 4;
  float4 v = (i < ud4) ? ((const float4*)ue)[i] : ((const float4*)ie)[i - ud4];
  ((float4*)A)[i] = v;
  ((float4*)B)[i] = make_float4(0.f, 0.f, 0.f, 0.f);
}

// ---------------- SpMM tile processor: wave-per-edge, lane owns 2 dims --------
__device__ __forceinline__ void process_tile(const int* __restrict__ s_r,
                                             const int* __restrict__ s_c,
                                             const float* __restrict__ s_v,
                                             int cnt,
                                             const float* __restrict__ egoP,
                                             float* __restrict__ accQ,
                                             int wave, int lane) {
  #pragma unroll 4
  for (int e = wave; e < cnt; e += 8) {
    const int   r = s_r[e];                 // same-address LDS read: broadcast
    const int   c = s_c[e];
    const float v = s_v[e];
    const float2 x = *(const float2*)(egoP + (long long)c * D_DIM + lane * 2);
    float* dst = accQ + (long long)r * D_DIM + lane * 2;
    __hip_atomic_fetch_add(dst + 0, v * x.x, __ATOMIC_RELAXED, __HIP_MEMORY_SCOPE_AGENT);
    __hip_atomic_fetch_add(dst + 1, v * x.y, __ATOMIC_RELAXED, __HIP_MEMORY_SCOPE_AGENT);
  }
}

// ------- kernel 2: SpMM scatter, TDM double-buffered edge staging -------------
__global__ void __launch_bounds__(256)
spmm_scatter(const float* __restrict__ egoP, float* __restrict__ accQ,
             const int* __restrict__ rows, const int* __restrict__ cols,
             const float* __restrict__ vals) {
  __shared__ int   s_rows[2][EDGE_TILE];
  __shared__ int   s_cols[2][EDGE_TILE];
  __shared__ float s_vals[2][EDGE_TILE];

  const int start0 = (blockIdx.x * TILES_PER_BLOCK) * EDGE_TILE;
  const int start1 = start0 + EDGE_TILE;
  int cnt0 = NNZ_CNT - start0; if (cnt0 > EDGE_TILE) cnt0 = EDGE_TILE;
  int cnt1 = NNZ_CNT - start1; if (cnt1 > EDGE_TILE) cnt1 = EDGE_TILE;

#if HAVE_TDM
  if (threadIdx.x == 0) {   // wave 0 issues all DMAs (TDM ignores EXEC)
    tdm_load_1d(lds_off(s_rows[0]), rows + start0, (unsigned)cnt0);
    tdm_load_1d(lds_off(s_cols[0]), cols + start0, (unsigned)cnt0);
    tdm_load_1d(lds_off(s_vals[0]), vals + start0, (unsigned)cnt0);
    if (cnt1 > 0) {
      tdm_load_1d(lds_off(s_rows[1]), rows + start1, (unsigned)cnt1);
      tdm_load_1d(lds_off(s_cols[1]), cols + start1, (unsigned)cnt1);
      tdm_load_1d(lds_off(s_vals[1]), vals + start1, (unsigned)cnt1);
      __builtin_amdgcn_s_wait_tensorcnt(3);   // tile0 landed (in-order), tile1 in flight
    } else {
      __builtin_amdgcn_s_wait_tensorcnt(0);
    }
  }
#else
  for (int t = threadIdx.x; t < cnt0; t += 256) {
    s_rows[0][t] = rows[start0 + t];
    s_cols[0][t] = cols[start0 + t];
    s_vals[0][t] = vals[start0 + t];
  }
  for (int t = threadIdx.x; t < cnt1; t += 256) {
    s_rows[1][t] = rows[start1 + t];
    s_cols[1][t] = cols[start1 + t];
    s_vals[1][t] = vals[start1 + t];
  }
#endif
  __syncthreads();

  const int wave = threadIdx.x >> 5;   // 8 waves / block
  const int lane = threadIdx.x & 31;

  process_tile(s_rows[0], s_cols[0], s_vals[0], cnt0, egoP, accQ, wave, lane);

  if (cnt1 > 0) {
#if HAVE_TDM
    if (threadIdx.x == 0) __builtin_amdgcn_s_wait_tensorcnt(0);
#endif
    __syncthreads();
    process_tile(s_rows[1], s_cols[1], s_vals[1], cnt1, egoP, accQ, wave, lane);
  }
}

// ---- kernel 3: perturb with normalized noise, zero next acc, fold outputs ----
__global__ void __launch_bounds__(256)
perturb_accum(float* __restrict__ q,            // in: spmm acc; out: ego_{k+1}
              float* __restrict__ p,            // zeroed for next layer's acc
              const float* __restrict__ nz_k,   // noise[k], (N, D)
              float* __restrict__ out_final,    // d_out[0 : N*D]
              float* __restrict__ out_cl,       // d_out[N*D : 2*N*D]
              const int k) {
  const int wave = threadIdx.x >> 5;            // one wave per row
  const int lane = threadIdx.x & 31;
  const long long row = (long long)blockIdx.x * 8 + wave;
  if (row >= N_CNT) return;
  const long long idx = row * D_DIM + lane * 2;

  const float2 nz = *(const float2*)(nz_k + idx);
  float ss = nz.x * nz.x + nz.y * nz.y;
  #pragma unroll
  for (int off = 16; off > 0; off >>= 1)        // wave32 butterfly reduce
    ss += __shfl_xor(ss, off, 32);
  const float scale = EPS_F / fmaxf(sqrtf(ss), 1e-12f);

  const float2 a = *(const float2*)(q + idx);
  float2 e;
  e.x = a.x + ((a.x > 0.f) ? 1.f : (a.x < 0.f ? -1.f : 0.f)) * nz.x * scale;
  e.y = a.y + ((a.y > 0.f) ? 1.f : (a.y < 0.f ? -1.f : 0.f)) * nz.y * scale;

  *(float2*)(q + idx) = e;                              // ego for next spmm
  *(float2*)(p + idx) = make_float2(0.f, 0.f);          // next accumulator = 0

  float2 acc;
  if (k == 0) {
    acc = e;
    *(float2*)(out_cl + idx) = e;                       // ego_cl (layer 1)
  } else {
    acc = *(const float2*)(out_final + idx);
    acc.x += e.x; acc.y += e.y;
    if (k == 2) { acc.x *= (1.f / 3.f); acc.y *= (1.f / 3.f); }
  }
  *(float2*)(out_final + idx) = acc;
}

// ------------------------------- launcher ------------------------------------
extern "C" void kernel_launch(void* const* d_in, const int* in_sizes, int n_in,
                              void* d_out, int out_size, void* d_ws, size_t ws_size,
                              hipStream_t stream) {
  const float* ue    = (const float*)d_in[0];
  const float* ie    = (const float*)d_in[1];
  const int*   rows  = (const int*)  d_in[2];
  const int*   cols  = (const int*)  d_in[3];
  const float* vals  = (const float*)d_in[4];
  const float* noise = (const float*)d_in[5];
  float* out = (float*)d_out;

  float* A = (float*)d_ws;      // ping-pong state buffers, N*D floats each
  float* B = A + ND;

  float* out_final = out;       // mean of layers -> [user_all | item_all]
  float* out_cl    = out + ND;  // layer-1 ego   -> [user_cl  | item_cl ]

  init_ego<<<(int)(ND / 4 / 256), 256, 0, stream>>>(ue, ie, A, B);

  const int edges_per_block = EDGE_TILE * TILES_PER_BLOCK;                 // 4096
  const int spmm_blocks = (NNZ_CNT + edges_per_block - 1) / edges_per_block; // 977
  const int pert_blocks = (N_CNT + 7) / 8;                                 // 18750

  float* P = A;  // ego input
  float* Q = B;  // zeroed accumulator
  for (int k = 0; k < 3; ++k) {
    spmm_scatter<<<spmm_blocks, 256, 0, stream>>>(P, Q, rows, cols, vals);
    perturb_accum<<<pert_blocks, 256, 0, stream>>>(Q, P, noise + (long long)k * ND,
                                                   out_final, out_cl, k);
    float* t = P; P = Q; Q = t;
  }
}